// activation_sub_up_64175401337015
// MI455X (gfx1250) — compile-verified
//
#include <hip/hip_runtime.h>

#define DIM 512          // columns in p (and in q)
#define C4  (DIM / 4)    // 128 float4 chunks per half-row

typedef __attribute__((ext_vector_type(4))) float f32x4;

static __device__ __forceinline__ float hw_tanh(float x) {
#if __has_builtin(__builtin_amdgcn_tanhf)
    return __builtin_amdgcn_tanhf(x);      // v_tanh_f32 (gfx1250 TRANS op)
#elif __has_builtin(__builtin_amdgcn_tanh_f32)
    return __builtin_amdgcn_tanh_f32(x);
#else
    return tanhf(x);                       // ocml fallback
#endif
}

// out[row, 0:512]    = p + a * tanh(q)
// out[row, 512:1024] = q
// One thread handles 4 consecutive columns of the q-half (and the matching
// 4 columns of the p-half): 2x b128 NT loads, 1x b128 cached load of `a`,
// 2x b128 NT stores. Fully coalesced: wave32 covers 512 contiguous bytes.
__global__ __launch_bounds__(256) void pq_tanh_fuse_kernel(
    const float* __restrict__ pq,
    const float* __restrict__ a,
    float* __restrict__ out,
    int n_chunks)   // rows * C4
{
    const int tid = blockIdx.x * blockDim.x + threadIdx.x;
    if (tid >= n_chunks) return;

    const int  row  = tid >> 7;          // / C4
    const int  col  = (tid & (C4 - 1)) << 2;   // float index within half-row
    const long base = (long)row * (2 * DIM);

    // Streamed, read-once data: non-temporal 128-bit loads
    const f32x4 p = __builtin_nontemporal_load((const f32x4*)(pq + base + col));
    const f32x4 q = __builtin_nontemporal_load((const f32x4*)(pq + base + DIM + col));
    // Reused broadcast vector: regular (cached) 128-bit load
    const f32x4 av = *(const f32x4*)(a + col);

    f32x4 r;
    r.x = __builtin_fmaf(av.x, hw_tanh(q.x), p.x);
    r.y = __builtin_fmaf(av.y, hw_tanh(q.y), p.y);
    r.z = __builtin_fmaf(av.z, hw_tanh(q.z), p.z);
    r.w = __builtin_fmaf(av.w, hw_tanh(q.w), p.w);

    // Write-once outputs: non-temporal 128-bit stores
    __builtin_nontemporal_store(r, (f32x4*)(out + base + col));
    __builtin_nontemporal_store(q, (f32x4*)(out + base + DIM + col));
}

extern "C" void kernel_launch(void* const* d_in, const int* in_sizes, int n_in,
                              void* d_out, int out_size, void* d_ws, size_t ws_size,
                              hipStream_t stream) {
    const float* pq  = (const float*)d_in[0];   // (rows, 2*DIM) f32
    const float* a   = (const float*)d_in[1];   // (DIM,) f32
    float*       out = (float*)d_out;           // (rows, 2*DIM) f32

    const int rows     = in_sizes[0] / (2 * DIM);
    const int n_chunks = rows * C4;             // one thread per 4 q-columns

    const int block = 256;                      // 8 wave32 per block
    const int grid  = (n_chunks + block - 1) / block;

    pq_tanh_fuse_kernel<<<grid, block, 0, stream>>>(pq, a, out, n_chunks);
}